// GaussianSplat3D_84112639525258
// MI455X (gfx1250) — compile-verified
//
#include <hip/hip_runtime.h>
#include <cmath>

#define NMAX 1024
#define MIN_COV 1.0e-4f
#define NEAR_PLANE 1.0e-4f
#define ALPHA_MAX 0.999f

typedef __attribute__((ext_vector_type(2))) float v2f;
typedef __attribute__((ext_vector_type(8))) float v8f;

// ws layout (float units):
//  [0*NMAX)             : key (unsorted)
//  [1..13]*NMAX         : unsorted SoA fields: ia ib ic mx my op minx maxx miny maxy cr cg cb
//  14*NMAX + g*4        : sorted prj[g]   = (ia, ib, ic, mx)      [float4, NMAX]
//  18*NMAX + g*4        : sorted box[g]   = (minx, maxx, miny, maxy)
//  22*NMAX + g*4        : sorted colop[g] = (cr, cg, cb, op)
//  26*NMAX + g          : sorted my[g]
//  27*NMAX              : int count of visible/overlapping gaussians

__global__ void gs_preprocess(const float* __restrict__ means,
                              const float* __restrict__ covs,
                              const float* __restrict__ cols,
                              const float* __restrict__ opac,
                              const float* __restrict__ intr,
                              const float* __restrict__ c2w,
                              int W, int H, int N, float* __restrict__ ws) {
    int n = blockIdx.x * blockDim.x + threadIdx.x;
    if (n == 0) *(int*)(ws + 27 * NMAX) = 0;
    if (n >= N) return;

    // R = C2W[:3,:3]^T ; t = -R @ C2W[:3,3]
    float R[3][3], t[3];
    #pragma unroll
    for (int i = 0; i < 3; ++i)
        #pragma unroll
        for (int j = 0; j < 3; ++j) R[i][j] = c2w[j * 4 + i];
    #pragma unroll
    for (int i = 0; i < 3; ++i)
        t[i] = -(R[i][0] * c2w[3] + R[i][1] * c2w[7] + R[i][2] * c2w[11]);

    float m0 = means[n * 3 + 0], m1 = means[n * 3 + 1], m2 = means[n * 3 + 2];
    float mc[3];
    #pragma unroll
    for (int i = 0; i < 3; ++i) mc[i] = R[i][0] * m0 + R[i][1] * m1 + R[i][2] * m2 + t[i];

    float Sv[3][3];
    #pragma unroll
    for (int i = 0; i < 3; ++i)
        #pragma unroll
        for (int j = 0; j < 3; ++j) Sv[i][j] = covs[n * 9 + i * 3 + j];

    // C = R S R^T
    float RS[3][3], C[3][3];
    #pragma unroll
    for (int i = 0; i < 3; ++i)
        #pragma unroll
        for (int k = 0; k < 3; ++k)
            RS[i][k] = R[i][0] * Sv[0][k] + R[i][1] * Sv[1][k] + R[i][2] * Sv[2][k];
    #pragma unroll
    for (int i = 0; i < 3; ++i)
        #pragma unroll
        for (int l = 0; l < 3; ++l)
            C[i][l] = RS[i][0] * R[l][0] + RS[i][1] * R[l][1] + RS[i][2] * R[l][2];

    float x = mc[0], y = mc[1], z = mc[2];
    bool vis = z > NEAR_PLANE;
    float sz = vis ? z : 1.0f;
    float fx = intr[0], fy = intr[4], cx = intr[2], cy = intr[5];
    float mx = fx * x / sz + cx;
    float my = fy * y / sz + cy;

    // J = [[u,0,wj],[0,v,sj]]
    float u = fx / sz, wj = -fx * x / (sz * sz);
    float v = fy / sz, sj = -fy * y / (sz * sz);
    float a = u * u * C[0][0] + u * wj * C[0][2] + wj * u * C[2][0] + wj * wj * C[2][2] + MIN_COV;
    float b = u * (C[0][1] * v + C[0][2] * sj) + wj * (C[2][1] * v + C[2][2] * sj);
    float c = v * v * C[1][1] + v * sj * C[1][2] + sj * v * C[2][1] + sj * sj * C[2][2] + MIN_COV;

    float disc = sqrtf(fmaxf((a - c) * (a - c) + 4.0f * b * b, 0.0f));
    float lam = fmaxf(0.5f * (a + c + disc), MIN_COV);
    float rad = 3.0f * sqrtf(lam);
    float mnx = floorf(mx - rad), mxx = ceilf(mx + rad);
    float mny = floorf(my - rad), mxy = ceilf(my + rad);
    bool ov = vis && (mxx >= 0.0f) && (mnx < (float)W) && (mxy >= 0.0f) && (mny < (float)H);

    float det = fmaxf(a * c - b * b, 1.0e-12f);
    float ia = c / det, ib = -b / det, ic = a / det;

    ws[0 * NMAX + n]  = ov ? z : __builtin_inff();
    ws[1 * NMAX + n]  = ia;
    ws[2 * NMAX + n]  = ib;
    ws[3 * NMAX + n]  = ic;
    ws[4 * NMAX + n]  = mx;
    ws[5 * NMAX + n]  = my;
    ws[6 * NMAX + n]  = ov ? opac[n] : 0.0f;
    ws[7 * NMAX + n]  = mnx;
    ws[8 * NMAX + n]  = mxx;
    ws[9 * NMAX + n]  = mny;
    ws[10 * NMAX + n] = mxy;
    ws[11 * NMAX + n] = cols[n * 3 + 0];
    ws[12 * NMAX + n] = cols[n * 3 + 1];
    ws[13 * NMAX + n] = cols[n * 3 + 2];
}

// One block of NMAX threads: stable rank sort by key, scatter into packed AoS.
__global__ void gs_ranksort(float* __restrict__ ws, int N) {
    __shared__ float sk[NMAX];
    int i = threadIdx.x;
    sk[i] = (i < N) ? ws[i] : __builtin_inff();
    __syncthreads();
    float ki = sk[i];
    int r = 0;
    for (int j = 0; j < NMAX; ++j) {
        float kj = sk[j];
        r += (kj < ki || (kj == ki && j < i)) ? 1 : 0;
    }
    if (i < N) {
        float4* prj   = (float4*)(ws + 14 * NMAX);
        float4* box   = (float4*)(ws + 18 * NMAX);
        float4* colop = (float4*)(ws + 22 * NMAX);
        float*  myv   = ws + 26 * NMAX;
        prj[r]   = make_float4(ws[1 * NMAX + i], ws[2 * NMAX + i],
                               ws[3 * NMAX + i], ws[4 * NMAX + i]);
        box[r]   = make_float4(ws[7 * NMAX + i], ws[8 * NMAX + i],
                               ws[9 * NMAX + i], ws[10 * NMAX + i]);
        colop[r] = make_float4(ws[11 * NMAX + i], ws[12 * NMAX + i],
                               ws[13 * NMAX + i], ws[6 * NMAX + i]);
        myv[r]   = ws[5 * NMAX + i];
        if (ki < __builtin_inff())
            atomicAdd((int*)(ws + 27 * NMAX), 1);
    }
}

__global__ __launch_bounds__(256) void gs_render(const float* __restrict__ ws,
                                                 int W, int H, int N,
                                                 float* __restrict__ out) {
    __shared__ float4 s_prj[NMAX];    // (ia, ib, ic, mx)
    __shared__ float4 s_box[NMAX];    // (minx, maxx, miny, maxy)
    __shared__ float4 s_colop[NMAX];  // (cr, cg, cb, op)
    __shared__ float  s_my[NMAX];

    const float4* g_prj   = (const float4*)(ws + 14 * NMAX);
    const float4* g_box   = (const float4*)(ws + 18 * NMAX);
    const float4* g_colop = (const float4*)(ws + 22 * NMAX);
    const float*  g_my    = ws + 26 * NMAX;

    int npad = (N + 15) & ~15;
    if (npad > NMAX) npad = NMAX;
    for (int i = threadIdx.x; i < npad; i += blockDim.x) {
        if (i < N) {
            s_prj[i]   = g_prj[i];
            s_box[i]   = g_box[i];
            s_colop[i] = g_colop[i];
            s_my[i]    = g_my[i];
        } else {
            s_prj[i]   = make_float4(0.0f, 0.0f, 0.0f, 0.0f);
            s_box[i]   = make_float4(3.0e38f, -3.0e38f, 3.0e38f, -3.0e38f);
            s_colop[i] = make_float4(0.0f, 0.0f, 0.0f, 0.0f);
            s_my[i]    = 0.0f;
        }
    }
    __syncthreads();

    int lane = threadIdx.x & 31;
    int wv = threadIdx.x >> 5;
    int strip = blockIdx.x * 8 + wv;
    int strips_x = (W + 15) >> 4;
    int total = strips_x * H;

    int M = *(const int*)(ws + 27 * NMAX);
    if (M < 0) M = 0;
    if (M > N) M = N;
    int nch = (M + 15) >> 4;

    if (strip < total) {
        int row = strip / strips_x;
        int tx0 = (strip - row * strips_x) << 4;
        int p = lane & 15;
        bool hi = lane >= 16;
        int px = tx0 + p;
        float fpx = (float)px, fpy = (float)row;
        float pl = (float)p;

        // B operand: K=0 -> px'^2, K=1 -> px', K=2 -> 1, K=3 -> 0
        v2f Bop;
        Bop.x = hi ? 1.0f : pl * pl;
        Bop.y = hi ? 0.0f : pl;

        float T = 1.0f, aR = 0.0f, aG = 0.0f, aB = 0.0f;
        v8f cz = 0.0f;

        for (int ch = 0; ch < nch; ++ch) {
            int g0 = ch << 4;
            int gA = g0 + p;
            float4 pj = s_prj[gA];
            float ia = pj.x, ibv = pj.y, icv = pj.z;
            float mxl = pj.w - (float)tx0;       // mx' in tile coords
            float myl = s_my[gA] - fpy;          // my - row  (dy = -my')
            float cB = -2.0f * (ia * mxl + ibv * myl);
            float cC = (ia * mxl + 2.0f * ibv * myl) * mxl + icv * myl * myl;
            v2f Aop;
            Aop.x = hi ? cC : ia;
            Aop.y = hi ? 0.0f : cB;

            // q[g,p] tile: D = A(16x4) x B(4x16), M=gaussian, N=pixel
            v8f q = __builtin_amdgcn_wmma_f32_16x16x4_f32(
                false, Aop, false, Bop, (short)0, cz, false, false);

            int gb = g0 + (hi ? 8 : 0);
            float al[8];
            #pragma unroll
            for (int r = 0; r < 8; ++r) {
                int g = gb + r;
                float4 bx = s_box[g];
                float op = s_colop[g].w;
                float wexp = __expf(-0.5f * q[r]);
                float av = fminf(op * wexp, ALPHA_MAX);
                // non-short-circuit: lowers to v_cmp + and + cndmask (branchless)
                bool ins = ((fpx >= bx.x) & (fpx <= bx.y)) &
                           ((fpy >= bx.z) & (fpy <= bx.w));
                al[r] = ins ? av : 0.0f;
            }

            // exchange halves so every lane has all 16 alphas in z-order
            float a16[16];
            #pragma unroll
            for (int r = 0; r < 8; ++r) {
                float o = __shfl_xor(al[r], 16, 32);
                a16[r]     = hi ? o : al[r];
                a16[r + 8] = hi ? al[r] : o;
            }

            #pragma unroll
            for (int k = 0; k < 16; ++k) {
                float4 co = s_colop[g0 + k];
                float av = a16[k];
                float wgt = T * av;
                aR += wgt * co.x;
                aG += wgt * co.y;
                aB += wgt * co.z;
                T -= T * av;
            }
        }

        if (!hi && px < W) {
            float* o = out + (row * W + px) * 3;
            o[0] = aR; o[1] = aG; o[2] = aB;
        }
    }
}

extern "C" void kernel_launch(void* const* d_in, const int* in_sizes, int n_in,
                              void* d_out, int out_size, void* d_ws, size_t ws_size,
                              hipStream_t stream) {
    const float* means = (const float*)d_in[0];
    const float* covs  = (const float*)d_in[1];
    const float* cols  = (const float*)d_in[2];
    const float* opac  = (const float*)d_in[3];
    const float* intr  = (const float*)d_in[4];
    const float* c2w   = (const float*)d_in[5];

    int N = in_sizes[0] / 3;
    if (N > NMAX) N = NMAX;
    int HW = out_size / 3;
    int Wd = (int)(sqrt((double)HW) + 0.5);
    if (Wd < 1) Wd = 1;
    int Hd = HW / Wd;

    float* ws = (float*)d_ws;
    float* img = (float*)d_out;

    gs_preprocess<<<(N + 255) / 256, 256, 0, stream>>>(means, covs, cols, opac,
                                                       intr, c2w, Wd, Hd, N, ws);
    gs_ranksort<<<1, NMAX, 0, stream>>>(ws, N);

    int strips_x = (Wd + 15) >> 4;
    int total = strips_x * Hd;
    int blocks = (total + 7) / 8;
    gs_render<<<blocks, 256, 0, stream>>>(ws, Wd, Hd, N, img);
}